// C3dLossKnnBtwnGT_15109694947702
// MI455X (gfx1250) — compile-verified
//
#include <hip/hip_runtime.h>

typedef float v2f __attribute__((ext_vector_type(2)));
typedef float v8f __attribute__((ext_vector_type(8)));

#define BB 2
#define PP 8192
#define KNN 20
#define QT 256          // queries per block
#define DT 32           // db points per tile
#define NT (PP / DT)    // 256 db tiles
#define NBLK (PP / QT)  // 32 blocks along queries
#define D2STRIDE 36     // 144 B rows: 16B-aligned b128 loads, conflict-free halves
#define IDXMASK 0x1FFFu         // 13 bits: P = 8192
#define KEYMASK 0xFFFFE000u     // truncated |d|^2 bits (order-preserving)
#define ELLC 0.15f
#define ELL_BASEC 10.0f
#define COLOR_SCALEC 0.2f
#define RES_MINC 0.1f
#define RES_MAXC 2.0f
#define BIGC 1e10f
#define EPSC 1e-12f

#if __has_builtin(__builtin_amdgcn_global_load_async_to_lds_b128) && \
    __has_builtin(__builtin_amdgcn_s_wait_asynccnt)
#define HAS_ASYNC_LDS 1
typedef int v4i_ __attribute__((ext_vector_type(4)));
typedef __attribute__((address_space(1))) v4i_ gv4;  // global-space 128-bit
typedef __attribute__((address_space(3))) v4i_ lv4;  // LDS-space 128-bit
#else
#define HAS_ASYNC_LDS 0
#endif

// ---------------------------------------------------------------------------
// Kernel 1: transform db clouds (R*x + t, R*n), precompute |p|^2 (+BIG mask)
// ---------------------------------------------------------------------------
__global__ void transform_db(const float* __restrict__ xyz1,
                             const float* __restrict__ xyz2,
                             const float* __restrict__ n1,
                             const float* __restrict__ n2,
                             const float* __restrict__ R12,
                             const float* __restrict__ t12,
                             const float* __restrict__ R21,
                             const float* __restrict__ t21,
                             const int* __restrict__ npts1,
                             const int* __restrict__ npts2,
                             float4* __restrict__ pdb,
                             float4* __restrict__ ndb) {
  int side = blockIdx.y;
  int tid = blockIdx.x * blockDim.x + threadIdx.x;
  if (tid >= BB * PP) return;
  int b = tid / PP, i = tid % PP;

  const float* src  = (side == 0) ? xyz2 : xyz1;
  const float* nsrc = (side == 0) ? n2   : n1;
  const float* R    = (side == 0) ? R12  : R21;
  const float* t    = (side == 0) ? t12  : t21;
  int lim           = (side == 0) ? npts2[b] : npts1[b];

  const float* p  = src  + (size_t)(b * PP + i) * 3;
  const float* nn = nsrc + (size_t)(b * PP + i) * 3;
  const float* Rb = R + b * 9;
  const float* tb = t + b * 3;

  float px = p[0], py = p[1], pz = p[2];
  float x = Rb[0] * px + Rb[1] * py + Rb[2] * pz + tb[0];
  float y = Rb[3] * px + Rb[4] * py + Rb[5] * pz + tb[1];
  float z = Rb[6] * px + Rb[7] * py + Rb[8] * pz + tb[2];
  float w = x * x + y * y + z * z + ((i < lim) ? 0.0f : BIGC);

  float ax = nn[0], ay = nn[1], az = nn[2];
  float nx = Rb[0] * ax + Rb[1] * ay + Rb[2] * az;
  float ny = Rb[3] * ax + Rb[4] * ay + Rb[5] * az;
  float nz = Rb[6] * ax + Rb[7] * ay + Rb[8] * az;

  size_t o = (size_t)side * BB * PP + (size_t)b * PP + i;
  pdb[o] = make_float4(x, y, z, w);
  ndb[o] = make_float4(nx, ny, nz, 0.0f);
}

// ---------------------------------------------------------------------------
// Kernel 2: fused WMMA distance tiles + per-query top-20 + loss terms.
// D = A*B + C with A = -2*q (16x4, K padded), B = p (4x16),
// C[M][N] = |q_M|^2 + |p_N|^2  ==>  D[M][N] = squared distance, one WMMA.
// d2 is packed with its db index into one u32 key (order-preserving, and
// low-index tie-breaking like jax top_k). db tiles are double-buffered in
// LDS via global_load_async_to_lds_b128 (ASYNCcnt pipeline).
// ---------------------------------------------------------------------------
__global__ __launch_bounds__(256) void knn_side(
    const float* __restrict__ xyzq, const float* __restrict__ hsvq,
    const float* __restrict__ nrmq, const float* __restrict__ resq,
    const int* __restrict__ nptsq,
    const float4* __restrict__ pdb, const float4* __restrict__ ndb,
    const float* __restrict__ hsvdb, const float* __restrict__ resdb,
    float* __restrict__ blockSums, int sumOffset) {
  __shared__ alignas(16) unsigned d2u[QT][D2STRIDE];
  __shared__ float qn_s[QT];
  __shared__ float red[QT];
#if HAS_ASYNC_LDS
  __shared__ alignas(16) float4 dbstage[2][DT];
#endif

  const int b = blockIdx.y;
  const int qbase = blockIdx.x * QT;
  const int t = threadIdx.x;
  const int qi = qbase + t;

  const int wave = t >> 5;
  const int lane = t & 31;
  const int lane16 = lane & 15;
  const int hi = lane >> 4;

  const float4* pd = pdb + (size_t)b * PP;

  // own query point; publish |q|^2
  const float* qp = xyzq + (size_t)(b * PP + qi) * 3;
  const float qx = qp[0], qy = qp[1], qz = qp[2];
  qn_s[t] = qx * qx + qy * qy + qz * qz;

#if HAS_ASYNC_LDS
  if (wave == 0) {  // stage tile 0 asynchronously (ASYNCcnt path)
    __builtin_amdgcn_global_load_async_to_lds_b128(
        (gv4*)(pd + lane), (lv4*)&dbstage[0][lane], 0, 0);
    __builtin_amdgcn_s_wait_asynccnt(0);
  }
#endif
  __syncthreads();  // publish qn_s (+ stage[0])

  // A fragments (-2*q) + row norms for this wave's two 16-row tiles.
  // 32-bit A 16x4 layout: lanes 0-15 hold K=0,1; lanes 16-31 hold K=2,3.
  v2f af[2];
  float rqn[2][8];
  for (int rt = 0; rt < 2; ++rt) {
    int row = (2 * wave + rt) * 16;
    const float* ap = xyzq + (size_t)(b * PP + qbase + row + lane16) * 3;
    float c0 = ap[0], c1 = ap[1], c2 = ap[2];
    af[rt].x = -2.0f * (hi ? c2 : c0);
    af[rt].y = -2.0f * (hi ? 0.0f : c1);
    for (int r = 0; r < 8; ++r) rqn[rt][r] = qn_s[row + r + hi * 8];
  }

  // per-query top-20 of packed keys (unsorted, threshold + replace-max)
  unsigned kk[KNN];
  for (int k = 0; k < KNN; ++k) kk[k] = 0xFFFFFFFFu;
  unsigned worstKey = 0xFFFFFFFFu;
  int wslot = 0;

  auto consider = [&](unsigned key) {
    if (key < worstKey) {
      kk[wslot] = key;
      unsigned w = 0;
      int ws = 0;
      for (int k = 0; k < KNN; ++k)
        if (kk[k] > w) { w = kk[k]; ws = k; }
      worstKey = w;
      wslot = ws;
    }
  };

  for (int tile = 0; tile < NT; ++tile) {
#if HAS_ASYNC_LDS
    if (wave == 0 && tile + 1 < NT) {  // overlap next tile's copy with WMMA
      __builtin_amdgcn_global_load_async_to_lds_b128(
          (gv4*)(pd + (tile + 1) * DT + lane),
          (lv4*)&dbstage[(tile + 1) & 1][lane], 0, 0);
    }
#else
    if (tile + 1 < NT) __builtin_prefetch(&pd[(tile + 1) * DT + lane], 0, 0);
#endif

    // B 4x16 layout mirrors A: lanes 0-15 hold K=0,1; lanes 16-31 K=2,3.
    for (int c = 0; c < 2; ++c) {
#if HAS_ASYNC_LDS
      float4 p = dbstage[tile & 1][c * 16 + lane16];
#else
      float4 p = pd[tile * DT + c * 16 + lane16];
#endif
      v2f bf;
      bf.x = hi ? p.z : p.x;
      bf.y = hi ? 0.0f : p.y;
      float pn = p.w;  // |p|^2 (+BIG for invalid db points)
      unsigned jcol = (unsigned)(tile * DT + c * 16 + lane16);
      for (int rt = 0; rt < 2; ++rt) {
        v8f cc;
        for (int r = 0; r < 8; ++r) cc[r] = rqn[rt][r] + pn;
        v8f d = __builtin_amdgcn_wmma_f32_16x16x4_f32(
            false, af[rt], false, bf, (short)0, cc, false, false);
        int rowb = (2 * wave + rt) * 16 + hi * 8;
        for (int r = 0; r < 8; ++r) {
          float dc = fmaxf(d[r], 0.0f);  // clamp: keep bit-order monotonic
          d2u[rowb + r][c * 16 + lane16] =
              (__float_as_uint(dc) & KEYMASK) | jcol;
        }
      }
    }
    __syncthreads();  // key tile visible to all

    // bulk-load this thread's row (8 x ds_load_b128); scan 4-wide, no outer
    // unroll so the replace body is duplicated only 4x (I-cache friendly)
    const uint4* row4 = (const uint4*)(&d2u[t][0]);
#pragma unroll 1
    for (int c4 = 0; c4 < DT / 4; ++c4) {
      uint4 v = row4[c4];
      consider(v.x);
      consider(v.y);
      consider(v.z);
      consider(v.w);
    }
#if HAS_ASYNC_LDS
    if (wave == 0 && tile + 1 < NT) __builtin_amdgcn_s_wait_asynccnt(0);
#endif
    __syncthreads();  // key tile reusable (+ next stage buffer published)
  }

  // loss terms for the 20 neighbors
  const float* hq = hsvq + (size_t)(b * PP + qi) * 3;
  const float* nq = nrmq + (size_t)(b * PP + qi) * 3;
  const float rq = resq[b * PP + qi];
  float lsl = fmaxf(ELLC * (qz - ELL_BASEC) / ELL_BASEC, ELLC);
  float ls = lsl * lsl;
  float h0 = hq[0], h1 = hq[1], h2 = hq[2];
  float nx = nq[0], ny = nq[1], nz = nq[2];
  const float4* nd = ndb + (size_t)b * PP;

  float acc = 0.0f;
  for (int k = 0; k < KNN; ++k) {
    unsigned key = kk[k];
    int j = (int)(key & IDXMASK);
    float d = __uint_as_float(key & KEYMASK);
    float distk = expf(-d / ls);
    const float* hk = hsvdb + (size_t)(b * PP + j) * 3;
    float dh0 = h0 - hk[0], dh1 = h1 - hk[1], dh2 = h2 - hk[2];
    float cd = sqrtf(dh0 * dh0 + dh1 * dh1 + dh2 * dh2 + EPSC);
    float colork = expf(-cd / COLOR_SCALEC);
    float4 nk = nd[j];
    float rk = resdb[b * PP + j];
    float alpha = (2.0f * RES_MINC) / (2.0f * RES_MINC / RES_MAXC + rq + rk);
    float dotn = nx * nk.x + ny * nk.y + nz * nk.z;
    acc += distk * colork * fmaxf(dotn * alpha, 0.0f);
  }
  if (qi >= nptsq[b]) acc = 0.0f;

  red[t] = acc;
  __syncthreads();
  for (int s = QT / 2; s > 0; s >>= 1) {
    if (t < s) red[t] += red[t + s];
    __syncthreads();
  }
  if (t == 0) blockSums[sumOffset + b * NBLK + blockIdx.x] = red[0];
}

// ---------------------------------------------------------------------------
// Kernel 3: deterministic ordered final reduction + normalization.
// ---------------------------------------------------------------------------
__global__ void finalize(const float* __restrict__ bs,
                         const int* __restrict__ npts1,
                         const int* __restrict__ npts2,
                         float* __restrict__ out) {
  if (threadIdx.x != 0 || blockIdx.x != 0) return;
  float s0 = 0.0f, s1 = 0.0f;
  for (int i = 0; i < BB * NBLK; ++i) s0 += bs[i];
  for (int i = 0; i < BB * NBLK; ++i) s1 += bs[BB * NBLK + i];
  float n1 = (float)(npts1[0] + npts1[1]);
  float n2 = (float)(npts2[0] + npts2[1]);
  float k1 = s0 / (n1 * (float)KNN);
  float k2 = s1 / (n2 * (float)KNN);
  out[0] = 0.5f * (k1 + k2);
}

extern "C" void kernel_launch(void* const* d_in, const int* in_sizes, int n_in,
                              void* d_out, int out_size, void* d_ws,
                              size_t ws_size, hipStream_t stream) {
  const float* xyz1 = (const float*)d_in[0];
  const float* xyz2 = (const float*)d_in[1];
  const float* hsv1 = (const float*)d_in[2];
  const float* hsv2 = (const float*)d_in[3];
  const float* normal1 = (const float*)d_in[4];
  const float* normal2 = (const float*)d_in[5];
  const float* nres1 = (const float*)d_in[6];
  const float* nres2 = (const float*)d_in[7];
  const float* R12 = (const float*)d_in[8];
  const float* t12 = (const float*)d_in[9];
  const float* R21 = (const float*)d_in[10];
  const float* t21 = (const float*)d_in[11];
  const int* npts1 = (const int*)d_in[12];
  const int* npts2 = (const int*)d_in[13];

  char* ws = (char*)d_ws;
  float4* pdb = (float4*)ws;                                   // 2*B*P float4
  float4* ndb = (float4*)(ws + (size_t)2 * BB * PP * sizeof(float4));
  float* blockSums = (float*)(ws + (size_t)4 * BB * PP * sizeof(float4));

  transform_db<<<dim3((BB * PP + 255) / 256, 2), 256, 0, stream>>>(
      xyz1, xyz2, normal1, normal2, R12, t12, R21, t21, npts1, npts2, pdb, ndb);

  // side 0: queries = cloud 1, db = cloud 2 transformed into frame 1
  knn_side<<<dim3(NBLK, BB), 256, 0, stream>>>(
      xyz1, hsv1, normal1, nres1, npts1,
      pdb, ndb, hsv2, nres2, blockSums, 0);
  // side 1: queries = cloud 2, db = cloud 1 transformed into frame 2
  knn_side<<<dim3(NBLK, BB), 256, 0, stream>>>(
      xyz2, hsv2, normal2, nres2, npts2,
      pdb + (size_t)BB * PP, ndb + (size_t)BB * PP, hsv1, nres1,
      blockSums, BB * NBLK);

  finalize<<<1, 32, 0, stream>>>(blockSums, npts1, npts2, (float*)d_out);
}